// MedianFilter_37563783971537
// MI455X (gfx1250) — compile-verified
//
#include <hip/hip_runtime.h>
#include <hip/hip_bf16.h>

// 3x3 median filter, zero-padded, fp32, NCHW (32,3,512,512).
// CDNA5 path: async global->LDS tile staging (ASYNCcnt) + LDS compute.

#define TW 128                 // output tile width
#define TH 16                  // output tile height
#define LDS_W 132              // (TW+2) padded to keep 16B-aligned rows (132*4 = 528B)
#define LDS_H (TH + 2)
#define IMG 512
#define NIMG (32 * 3)

typedef __attribute__((ext_vector_type(4))) float f32x4;

__global__ __launch_bounds__(256) void median3x3_kernel(const float* __restrict__ x,
                                                        float* __restrict__ out) {
    __shared__ __align__(16) float tile_s[LDS_H * LDS_W];

    const int tid    = threadIdx.x;
    const int tile_x = blockIdx.x;       // 0..3
    const int tile_y = blockIdx.y;       // 0..31
    const int img    = blockIdx.z;       // 0..95
    const int gx0 = tile_x * TW - 1;     // tile col c -> global x = gx0 + c
    const int gy0 = tile_y * TH - 1;     // tile row r -> global y = gy0 + r
    const float* base = x + (size_t)img * IMG * IMG;

    // ---- Stage (TH+2)x(TW+2) halo tile into LDS with CDNA5 async loads ----
    // Out-of-image cells get 0.0f (matches the reference's implicit zero pad).
    for (int i = tid; i < LDS_H * (TW + 2); i += 256) {
        const int r  = i / (TW + 2);
        const int c  = i - r * (TW + 2);
        const int gy = gy0 + r;
        const int gx = gx0 + c;
        const int lidx = r * LDS_W + c;
        if ((unsigned)gy < (unsigned)IMG && (unsigned)gx < (unsigned)IMG) {
            const float* gp = base + (size_t)gy * IMG + gx;
            // Flat address of an LDS object carries the LDS byte offset in
            // its low 32 bits (ISA aperture layout) -> valid VDST for async.
            unsigned ldsa = (unsigned)(unsigned long long)(&tile_s[lidx]);
            asm volatile("global_load_async_to_lds_b32 %0, %1, off"
                         :: "v"(ldsa), "v"(gp) : "memory");
        } else {
            tile_s[lidx] = 0.0f;
        }
    }
    // ASYNCcnt is per-wave: each wave drains its own async loads, then the
    // workgroup barrier makes the LDS tile visible to all waves.
    asm volatile("s_wait_asynccnt 0" ::: "memory");
    __syncthreads();

    // ---- Compute: each thread does a 4(w) x 2(h) output patch ----
    const int tx = tid & 31;             // 0..31 -> column group
    const int ty = tid >> 5;             // 0..7  -> row pair
    const int ox = tx << 2;              // 0..124 (local out x)
    const int oy = ty << 1;              // 0..14  (local out y)

    // v[r][c]: input rows oy..oy+3, cols ox..ox+5 (tile coords).
    float v[4][6];
    #pragma unroll
    for (int r = 0; r < 4; ++r) {
        const int off = (oy + r) * LDS_W + ox;           // 16B aligned
        f32x4 q = *(const f32x4*)&tile_s[off];           // ds_load_b128
        v[r][0] = q.x; v[r][1] = q.y; v[r][2] = q.z; v[r][3] = q.w;
        v[r][4] = tile_s[off + 4];
        v[r][5] = tile_s[off + 5];
    }

    float* o0 = out + (size_t)img * IMG * IMG
                    + (size_t)(tile_y * TH + oy) * IMG
                    + (size_t)(tile_x * TW + ox);

    #pragma unroll
    for (int dy = 0; dy < 2; ++dy) {
        // Sort each of the 6 shared columns (3 vertical elements each):
        // 6 min/max ops per column.
        float lo[6], mi[6], hi[6];
        #pragma unroll
        for (int c = 0; c < 6; ++c) {
            float a = v[dy][c], b = v[dy + 1][c], d = v[dy + 2][c];
            float mn = fminf(a, b);
            float mx = fmaxf(a, b);
            lo[c] = fminf(mn, d);
            hi[c] = fmaxf(mx, d);
            mi[c] = fmaxf(mn, fminf(mx, d));
        }
        // Exact identity: med9 = med3( max3(col mins), med3(col meds), min3(col maxes) )
        f32x4 res;
        #pragma unroll
        for (int j = 0; j < 4; ++j) {
            float maxlo = fmaxf(fmaxf(lo[j], lo[j + 1]), lo[j + 2]);
            float minhi = fminf(fminf(hi[j], hi[j + 1]), hi[j + 2]);
            float a = mi[j], b = mi[j + 1], d = mi[j + 2];
            float medmi = fmaxf(fminf(a, b), fminf(fmaxf(a, b), d));
            float m = fmaxf(fminf(maxlo, medmi),
                            fminf(fmaxf(maxlo, medmi), minhi));
            if (j == 0) res.x = m;
            else if (j == 1) res.y = m;
            else if (j == 2) res.z = m;
            else res.w = m;
        }
        // Write-once output: non-temporal 128-bit store, keep L2 for input reuse.
        __builtin_nontemporal_store(res, (f32x4*)(o0 + (size_t)dy * IMG));
    }
}

extern "C" void kernel_launch(void* const* d_in, const int* in_sizes, int n_in,
                              void* d_out, int out_size, void* d_ws, size_t ws_size,
                              hipStream_t stream) {
    const float* x = (const float*)d_in[0];
    float* out = (float*)d_out;
    dim3 grid(IMG / TW, IMG / TH, NIMG);   // (4, 32, 96)
    median3x3_kernel<<<grid, dim3(256), 0, stream>>>(x, out);
}